// DandCrecon_50105088475769
// MI455X (gfx1250) — compile-verified
//
#include <hip/hip_runtime.h>
#include <cstdint>
#include <math.h>

// Problem constants (from reference)
constexpr int cA3 = 180, cA2 = 128, cW = 384;   // sinogram [A3, A2, W2D]
constexpr int cNV = 192, cNU = 256;             // projection [A3, NV, NU]
constexpr int cVZ = 192, cVY = 192, cVX = 192;  // volume
constexpr int PNU = 272;                        // padded proj row: 8 | 256 | 8
constexpr float PI_F  = 3.14159265358979323846f;
constexpr float DSD_F = 1200.0f;
constexpr float DSO_F = 750.0f;

typedef __attribute__((ext_vector_type(2))) float v2f;
typedef __attribute__((ext_vector_type(8))) float v8f;
typedef int v4i __attribute__((vector_size(16)));
typedef __attribute__((address_space(1))) v4i* gptr4;   // global b128
typedef __attribute__((address_space(3))) v4i* lptr4;   // LDS b128

// ---------------------------------------------------------------------------
// gfx1250 async global->LDS helpers (builtin if present, else inline asm)
// ---------------------------------------------------------------------------
__device__ __forceinline__ void async_copy_b128(const float* gsrc, float* lds_dst) {
#if defined(__has_builtin) && __has_builtin(__builtin_amdgcn_global_load_async_to_lds_b128)
  __builtin_amdgcn_global_load_async_to_lds_b128(
      (gptr4)(uintptr_t)gsrc, (lptr4)(uintptr_t)lds_dst, 0, 0);
#else
  uint32_t ldsb = (uint32_t)(uintptr_t)lds_dst;
  uint64_t g    = (uint64_t)(uintptr_t)gsrc;
  asm volatile("global_load_async_to_lds_b128 %0, %1, off"
               :: "v"(ldsb), "v"(g) : "memory");
#endif
}

__device__ __forceinline__ void wait_async0() {
#if defined(__has_builtin) && __has_builtin(__builtin_amdgcn_s_wait_asynccnt)
  __builtin_amdgcn_s_wait_asynccnt(0);
#else
  asm volatile("s_wait_asynccnt 0" ::: "memory");
#endif
}

// Branchless stencil coefficient G[k][n] (cndmask chains, no control flow).
__device__ __forceinline__ float gcoef(int k, int n) {
  float v = (k == n - 1) ? -0.5f : ((k == n + 1) ? 0.5f : 0.0f);
  v = (n == 0)      ? ((k == 0)      ? -1.0f : ((k == 1)      ? 1.0f : 0.0f)) : v;
  v = (n == cW - 1) ? ((k == cW - 2) ? -1.0f : ((k == cW - 1) ? 1.0f : 0.0f)) : v;
  return v;
}

// ---------------------------------------------------------------------------
// Stage 1: ws = sino*weight, d = grad_last(ws) == d = ws @ G (banded 384x384)
// as V_WMMA_F32_16X16X4_F32 over the nonzero band. LDS rows are zero-padded
// (k in [-4, 388)) so A-fragment loads are unconditional ds_load_b64; row
// stride 452 (== 4 mod 64) -> conflict-free and 8B-aligned (k0 always even).
// ---------------------------------------------------------------------------
__global__ void k_grad_wmma(const float* __restrict__ sino,
                            const float* __restrict__ wgt,
                            float* __restrict__ dgrad) {
  constexpr int WP = 452;                    // padded LDS row stride
  extern __shared__ float spanel[];          // 128 * WP floats
  const int tid  = threadIdx.x;              // blockDim = 256 (8 waves)
  const size_t base = (size_t)blockIdx.x * 128 * cW;

  // Zero the pad cells (k = -4..-1 -> idx 0..3, k = 384..387 -> idx 388..391).
  for (int i = tid; i < 128 * 8; i += 256) {
    int r = i >> 3, p = i & 7;
    spanel[r * WP + (p < 4 ? p : 384 + p)] = 0.0f;
  }
  // Coalesced staging of 128 contiguous rows of (sino * weight).
  for (int i = tid; i < 128 * cW; i += 256) {
    int r = i / cW, c = i - r * cW;
    spanel[r * WP + 4 + c] = sino[base + i] * wgt[base + i];
  }
  __syncthreads();

  const int wv   = tid >> 5;                 // wave id: 16-row panel in block
  const int lane = tid & 31;
  const int m    = lane & 15;                // M index within 16x16 tile
  const int kb   = (lane >> 4) * 2;          // K sub-offset per A-layout
  const int hi8  = (lane >> 4) * 8;          // D-layout row offset
  const float* prow = spanel + (wv * 16 + m) * WP + 4;   // points at k = 0
  const size_t orow = base + (size_t)(wv * 16) * cW;

  // B fragments: interior tiles share one set (depends on k-n only);
  // tiles n=0 and n=23 get edge-patched sets. All built branchlessly.
  v2f Bint[6], B0[6], B23[6];
#pragma unroll
  for (int j = 0; j < 6; ++j) {
    const int k0 = -4 + 4 * j + kb;          // K offset relative to w0
    const int d0 = k0 - m, d1 = d0 + 1;
    Bint[j].x = (d0 == -1) ? -0.5f : ((d0 == 1) ? 0.5f : 0.0f);
    Bint[j].y = (d1 == -1) ? -0.5f : ((d1 == 1) ? 0.5f : 0.0f);
    B0[j].x  = gcoef(k0, m);             B0[j].y  = gcoef(k0 + 1, m);
    B23[j].x = gcoef(368 + k0, 368 + m); B23[j].y = gcoef(368 + k0 + 1, 368 + m);
  }

  auto do_tile = [&](int w0, const v2f* B) {
    v8f acc = {};
#pragma unroll
    for (int j = 0; j < 6; ++j) {
      const float2 af = *(const float2*)(prow + (w0 - 4 + 4 * j + kb));
      v2f a; a.x = af.x; a.y = af.y;
      acc = __builtin_amdgcn_wmma_f32_16x16x4_f32(
          false, a, false, B[j], (short)0, acc, false, false);
    }
#pragma unroll
    for (int i = 0; i < 8; ++i)              // D: 2 rows x 16 cols per VGPR
      dgrad[orow + (size_t)(i + hi8) * cW + w0 + m] = acc[i];
  };

  do_tile(0, B0);
  for (int n = 1; n < 23; ++n) do_tile(n * 16, Bint);
  do_tile(368, B23);
}

// ---------------------------------------------------------------------------
// Stage 2: 2D parallel-beam BP + cosine detector weight.
// One block = (one a3 slice, quarter of the v rows); the 192 KB slice is
// DMA'd to LDS via async b128 loads. Bounds masks dropped (provably inactive:
// |pos-191.5| <= sqrt(255^2+191^2)/2 ~ 159.3 -> i0 in [32, 350]).
// Loop nest: angle OUTER, 24 v-rows per thread in registers -> trig load
// amortized 24x, 24 independent accumulators hide LDS gather latency.
// Writes proj with padded row stride PNU=272 (zeroed borders for stage 3).
// ---------------------------------------------------------------------------
__global__ void k_bp2d(const float* __restrict__ dgrad,
                       float* __restrict__ proj) {
  extern __shared__ float lds[];             // cA2*cW data + 2*cA2 trig
  float* sdata = lds;                        // 49152 floats
  float* strig = lds + cA2 * cW;             // 256 floats (c*0.5, s*0.5)
  const int tid = threadIdx.x;               // blockDim = 512
  const int a3 = blockIdx.x;
  const int chunk = blockIdx.y;              // 0..3 -> 48 v-rows each
  const float* slice = dgrad + (size_t)a3 * (cA2 * cW);

  // Kick off async DMA of the slice (12288 x b128).
  for (int i = tid; i < (cA2 * cW) / 4; i += 512)
    async_copy_b128(slice + i * 4, sdata + i * 4);

  // Overlap with the DMA: per-angle trig + zero the proj row borders.
  if (tid < cA2) {
    float th = (float)tid * (PI_F / cA2);
    float s, c;
    sincosf(th, &s, &c);
    strig[2 * tid]     = 0.5f * c;           // pre-divided by DS2D=2
    strig[2 * tid + 1] = 0.5f * s;
  }
  for (int q = tid; q < 48 * 16; q += 512) {
    int rv  = chunk * 48 + (q >> 4);
    int pp  = q & 15;
    int col = (pp < 8) ? pp : (256 + pp);    // 0..7 and 264..271
    proj[((size_t)a3 * cNV + rv) * PNU + col] = 0.0f;
  }
  wait_async0();
  __syncthreads();

  const int ui = tid & 255;                  // one detector column per thread
  const int rg = tid >> 8;                   // row group 0/1 (24 rows each)
  const int vbase = chunk * 48 + rg * 24;
  const float X = ((float)ui - 127.5f) * 2.0f;

  float acc[24];
#pragma unroll
  for (int r = 0; r < 24; ++r) acc[r] = 0.0f;

  for (int a = 0; a < cA2; ++a) {
    float2 cs = ((const float2*)strig)[a];   // broadcast ds_load_b64
    const float* row = sdata + a * cW;
    const float posb = fmaf(X, cs.x, 191.5f);
#pragma unroll
    for (int r = 0; r < 24; ++r) {
      const float Yr = ((float)(vbase + r) - 95.5f) * 2.0f;
      float pos = fmaf(Yr, cs.y, posb);
      float fp  = floorf(pos);
      float f   = pos - fp;
      const float* q = row + (int)fp;        // always in [32, 350]
      acc[r] = fmaf(1.0f - f, q[0], acc[r]);
      acc[r] = fmaf(f,        q[1], acc[r]);
    }
  }

  const float norm = PI_F / cA2;
#pragma unroll
  for (int r = 0; r < 24; ++r) {
    const int vi = vbase + r;
    const float Yr = ((float)vi - 95.5f) * 2.0f;
    float wdet = DSD_F / sqrtf(fmaf(X, X, fmaf(Yr, Yr, DSD_F * DSD_F)));
    proj[((size_t)a3 * cNV + vi) * PNU + 8 + ui] = acc[r] * norm * wdet;
  }
}

// ---------------------------------------------------------------------------
// Stage 3: cone-beam BP (dominant stage). 4 z-voxels per thread: per-view
// geometry (r, t, 1/denom, iu, fu, q, w^2) is z-invariant and amortized 4x.
// iv in [2.3, 188.7] -> no v masks; iu in [-4.4, 259.4] -> handled by the
// zeroed 8-column borders of proj. All taps are unconditional loads.
// ---------------------------------------------------------------------------
__global__ void k_bp3d(const float* __restrict__ proj,
                       float* __restrict__ out) {
  __shared__ float trig[2 * cA3];
  const int flat = threadIdx.x + 64 * threadIdx.y + 128 * threadIdx.z;
  if (flat < cA3) {
    float b = (float)flat * (2.0f * PI_F / cA3);
    float s, c;
    sincosf(b, &s, &c);
    trig[2 * flat]     = c;
    trig[2 * flat + 1] = s;
  }
  __syncthreads();

  const int xi = blockIdx.x * 64 + threadIdx.x;
  const int yi = blockIdx.y * 2 + threadIdx.y;
  const int z0 = blockIdx.z * 8 + threadIdx.z * 4;   // 4 z per thread
  const float X  = (float)xi - 95.5f;
  const float Y  = (float)yi - 95.5f;
  const float Zb = (float)z0 - 95.5f;

  float acc[4] = {0.0f, 0.0f, 0.0f, 0.0f};

  for (int a = 0; a < cA3; ++a) {
    float2 cs = ((const float2*)trig)[a];
    const float cb = cs.x, sb = cs.y;
    float r  = fmaf(X, cb, Y * sb);
    float t  = fmaf(-X, sb, Y * cb);
    float rd = 1.0f / (DSO_F - r);
    float q  = 600.0f * rd;                  // DSD/DU = DSD/DV = 600
    float iu = fmaf(t, q, 127.5f);
    float fu0 = floorf(iu);
    float fu  = iu - fu0;
    float wu0 = 1.0f - fu;
    const int i0u = (int)fu0;                // in [-5, 259]
    const float* base = proj + (size_t)a * (cNV * PNU) + 8 + i0u;
    float w  = DSO_F * rd;
    float w2 = w * w;
#pragma unroll
    for (int zz = 0; zz < 4; ++zz) {
      float iv  = fmaf(Zb + (float)zz, q, 95.5f);
      float fv0 = floorf(iv);
      float fv  = iv - fv0;
      const float* r0 = base + (int)fv0 * PNU;     // i0v in [2, 188]
      float hi = fmaf(wu0, r0[0],   fu * r0[1]);
      float lo = fmaf(wu0, r0[PNU], fu * r0[PNU + 1]);
      float val = fmaf(fv, lo - hi, hi);           // (1-fv)*hi + fv*lo
      acc[zz] = fmaf(w2, val, acc[zz]);
    }
  }

  const float s = 2.0f * PI_F / cA3;
#pragma unroll
  for (int zz = 0; zz < 4; ++zz)
    out[((size_t)(z0 + zz) * cVY + yi) * cVX + xi] = acc[zz] * s;
}

// ---------------------------------------------------------------------------
extern "C" void kernel_launch(void* const* d_in, const int* in_sizes, int n_in,
                              void* d_out, int out_size, void* d_ws, size_t ws_size,
                              hipStream_t stream) {
  (void)in_sizes; (void)n_in; (void)out_size; (void)ws_size;
  const float* sino = (const float*)d_in[0];      // [1,1,180,128,384] f32
  const float* wgt  = (const float*)d_in[1];      // [180,128,384]     f32
  float* out = (float*)d_out;                     // [1,192,192,192]   f32

  float* dgrad = (float*)d_ws;                               // 35.4 MB
  float* proj  = dgrad + (size_t)cA3 * cA2 * cW;             // 37.6 MB (padded)

  // Stage 1: gradient as banded WMMA matmul (1440 16-row panels exactly).
  {
    size_t lds = (size_t)128 * 452 * sizeof(float);          // 231424 B
    k_grad_wmma<<<dim3(cA3), dim3(256), lds, stream>>>(sino, wgt, dgrad);
  }

  // Stage 2: 2D parallel BP, slice staged to LDS via async b128 DMA.
  {
    size_t lds = (size_t)(cA2 * cW + 2 * cA2) * sizeof(float); // 197632 B
    k_bp2d<<<dim3(cA3, 4), dim3(512), lds, stream>>>(dgrad, proj);
  }

  // Stage 3: cone-beam BP, 4 z-voxels per thread.
  k_bp3d<<<dim3(cVX / 64, cVY / 2, cVZ / 8), dim3(64, 2, 2), 0, stream>>>(proj, out);
}